// GATGNN_6691559047586
// MI455X (gfx1250) — compile-verified
//
#include <hip/hip_runtime.h>

#define NN 50000
#define EE 800000
#define CC 64
#define EDD 11
#define NEGS 0.2f

typedef __attribute__((ext_vector_type(2))) float v2f;
typedef __attribute__((ext_vector_type(8))) float v8f;

__device__ __forceinline__ void atomicMaxF(float* addr, float v) {
    // works for any mix of signs given init = -inf
    if (v >= 0.0f) atomicMax((int*)addr, __float_as_int(v));
    else           atomicMin((unsigned int*)addr, __float_as_uint(v));
}

// ---------------- K0: init out = x + bias; m = -inf; denom = 0; mean acc = 0
__global__ void k_init(const float* __restrict__ x, const float* __restrict__ bias,
                       float* __restrict__ out, float* __restrict__ mmax,
                       float* __restrict__ denom, float* __restrict__ macc) {
    int i = blockIdx.x * blockDim.x + threadIdx.x;
    if (i < NN * CC) out[i] = x[i] + bias[i & (CC - 1)];
    if (i < NN) { mmax[i] = -__builtin_inff(); denom[i] = 0.0f; }
    if (i < 16) macc[i] = 0.0f;
}

// ---------------- K1: column sums of edge_attr (mean computed at use site)
__global__ void k_meansum(const float* __restrict__ ea, float* __restrict__ macc) {
    __shared__ float s[EDD];
    if (threadIdx.x < EDD) s[threadIdx.x] = 0.0f;
    __syncthreads();
    float loc[EDD];
#pragma unroll
    for (int k = 0; k < EDD; k++) loc[k] = 0.0f;
    for (int e = blockIdx.x * blockDim.x + threadIdx.x; e < EE; e += gridDim.x * blockDim.x) {
#pragma unroll
        for (int k = 0; k < EDD; k++) loc[k] += ea[e * EDD + k];
    }
#pragma unroll
    for (int k = 0; k < EDD; k++) atomicAdd(&s[k], loc[k]);
    __syncthreads();
    if (threadIdx.x < EDD) atomicAdd(&macc[threadIdx.x], s[threadIdx.x]);
}

// ---------------- K2: xl = x@Wl + bl, xr = x@Wr + br via V_WMMA_F32_16X16X4_F32
// One wave owns a 16-row block; 4 column tiles per matrix; K=64 in steps of 4.
__global__ void k_proj_wmma(const float* __restrict__ x,
                            const float* __restrict__ Wl, const float* __restrict__ bl,
                            const float* __restrict__ Wr, const float* __restrict__ br,
                            float* __restrict__ xl, float* __restrict__ xr) {
    const int wid = blockIdx.x * (blockDim.x >> 5) + (threadIdx.x >> 5);
    if (wid >= NN / 16) return;                 // whole-wave uniform exit: EXEC stays all-ones
    const int lane = threadIdx.x & 31;
    const int half = lane >> 4;                 // 0: K pair {0,1}, 1: K pair {2,3}
    const int l    = lane & 15;
    const int m0   = wid * 16;

    v8f accL[4] = {};
    v8f accR[4] = {};

    for (int k0 = 0; k0 < CC; k0 += 4) {
        // A fragment (16x4): lanes 0-15 hold M=l K={k0,k0+1}; lanes 16-31 K={k0+2,k0+3}
        v2f a;
        const float* ap = x + (m0 + l) * CC + k0 + 2 * half;
        a.x = ap[0];
        a.y = ap[1];
#pragma unroll
        for (int n = 0; n < 4; n++) {
            const int kb = (k0 + 2 * half) * CC + n * 16 + l;
            v2f b;
            b.x = Wl[kb];
            b.y = Wl[kb + CC];
            accL[n] = __builtin_amdgcn_wmma_f32_16x16x4_f32(
                false, a, false, b, (short)0, accL[n], false, false);
            v2f c;
            c.x = Wr[kb];
            c.y = Wr[kb + CC];
            accR[n] = __builtin_amdgcn_wmma_f32_16x16x4_f32(
                false, a, false, c, (short)0, accR[n], false, false);
        }
    }

    // C/D layout: VGPR r -> row (half*8 + r), col = n*16 + l
#pragma unroll
    for (int n = 0; n < 4; n++) {
#pragma unroll
        for (int r = 0; r < 8; r++) {
            const int row = m0 + half * 8 + r;
            const int col = n * 16 + l;
            xl[row * CC + col] = accL[n][r] + bl[col];
            xr[row * CC + col] = accR[n][r] + br[col];
        }
    }
}

// ---------------- K3: per-edge attention logits + segment max
// One wave per edge, 2 channels per lane. W_e/att/mean cached in LDS.
__global__ void k_logits(const int* __restrict__ eidx, const float* __restrict__ ea,
                         const float* __restrict__ We, const float* __restrict__ att,
                         const float* __restrict__ macc,
                         const float* __restrict__ xl, const float* __restrict__ xr,
                         float* __restrict__ lg, float* __restrict__ mmax) {
    __shared__ float sWe[EDD * CC];
    __shared__ float sAtt[CC];
    __shared__ float sMean[EDD];
    for (int i = threadIdx.x; i < EDD * CC; i += blockDim.x) sWe[i] = We[i];
    if (threadIdx.x < CC)  sAtt[threadIdx.x]  = att[threadIdx.x];
    if (threadIdx.x < EDD) sMean[threadIdx.x] = macc[threadIdx.x] * (1.0f / (float)EE);
    __syncthreads();

    const int lane = threadIdx.x & 31;
    const int c0   = lane * 2;
    const int nw   = (gridDim.x * blockDim.x) >> 5;

    for (int e = (blockIdx.x * blockDim.x + threadIdx.x) >> 5; e < EE + NN; e += nw) {
        int s, d;
        float myea;
        if (e < EE) {
            s = eidx[e];
            d = eidx[EE + e];
            myea = (lane < EDD) ? ea[e * EDD + lane] : 0.0f;
        } else {
            s = e - EE;
            d = s;
            myea = (lane < EDD) ? sMean[lane] : 0.0f;
        }
        const float2 vl = *(const float2*)(xl + s * CC + c0);
        const float2 vr = *(const float2*)(xr + d * CC + c0);
        float hx = vl.x + vr.x;
        float hy = vl.y + vr.y;
#pragma unroll
        for (int k = 0; k < EDD; k++) {
            const float ek = __shfl(myea, k, 32);
            hx += ek * sWe[k * CC + c0];
            hy += ek * sWe[k * CC + c0 + 1];
        }
        hx = hx > 0.0f ? hx : NEGS * hx;
        hy = hy > 0.0f ? hy : NEGS * hy;
        float p = hx * sAtt[c0] + hy * sAtt[c0 + 1];
#pragma unroll
        for (int off = 16; off > 0; off >>= 1) p += __shfl_xor(p, off, 32);
        if (lane == 0) {
            lg[e] = p;
            atomicMaxF(&mmax[d], p);
        }
    }
}

// ---------------- K4: p = exp(logit - m[dst]); denom[dst] += p  (in-place over lg)
__global__ void k_expsum(const int* __restrict__ eidx, const float* __restrict__ mmax,
                         float* __restrict__ lg, float* __restrict__ denom) {
    int e = blockIdx.x * blockDim.x + threadIdx.x;
    if (e >= EE + NN) return;
    const int d = (e < EE) ? eidx[EE + e] : (e - EE);
    const float p = expf(lg[e] - mmax[d]);
    lg[e] = p;
    atomicAdd(&denom[d], p);
}

// ---------------- K5: out[dst] += (p/denom[dst]) * xl[src]  (32 lanes x 2ch per edge)
__global__ void k_scatter(const int* __restrict__ eidx, const float* __restrict__ lg,
                          const float* __restrict__ denom, const float* __restrict__ xl,
                          float* __restrict__ out) {
    int idx = blockIdx.x * blockDim.x + threadIdx.x;
    int e = idx >> 5;
    if (e >= EE + NN) return;
    const int lane = idx & 31;
    int s, d;
    if (e < EE) { s = eidx[e]; d = eidx[EE + e]; }
    else        { s = e - EE;  d = s; }
    const float alpha = lg[e] / denom[d];
    const int c0 = lane * 2;
    const float2 v = *(const float2*)(xl + s * CC + c0);
    atomicAdd(&out[d * CC + c0],     alpha * v.x);
    atomicAdd(&out[d * CC + c0 + 1], alpha * v.y);
}

extern "C" void kernel_launch(void* const* d_in, const int* in_sizes, int n_in,
                              void* d_out, int out_size, void* d_ws, size_t ws_size,
                              hipStream_t stream) {
    const float* x    = (const float*)d_in[0];
    const int*   eidx = (const int*)  d_in[1];
    const float* ea   = (const float*)d_in[2];
    const float* Wl   = (const float*)d_in[3];
    const float* bl   = (const float*)d_in[4];
    const float* Wr   = (const float*)d_in[5];
    const float* br   = (const float*)d_in[6];
    const float* We   = (const float*)d_in[7];
    const float* att  = (const float*)d_in[8];
    const float* bias = (const float*)d_in[9];
    float* out = (float*)d_out;

    float* ws    = (float*)d_ws;
    float* xl    = ws;                  // N*C
    float* xr    = xl + NN * CC;        // N*C
    float* lg    = xr + NN * CC;        // E+N  (logits, then p)
    float* mmax  = lg + (EE + NN);      // N
    float* denom = mmax + NN;           // N
    float* macc  = denom + NN;          // 16

    const int T = 256;
    k_init<<<(NN * CC + T - 1) / T, T, 0, stream>>>(x, bias, out, mmax, denom, macc);
    k_meansum<<<512, T, 0, stream>>>(ea, macc);
    k_proj_wmma<<<(NN / 16 + 7) / 8, T, 0, stream>>>(x, Wl, bl, Wr, br, xl, xr);
    k_logits<<<2048, T, 0, stream>>>(eidx, ea, We, att, macc, xl, xr, lg, mmax);
    k_expsum<<<(EE + NN + T - 1) / T, T, 0, stream>>>(eidx, mmax, lg, denom);
    k_scatter<<<((EE + NN) * 32 + T - 1) / T, T, 0, stream>>>(eidx, lg, denom, xl, out);
}